// Group_66700842107503
// MI455X (gfx1250) — compile-verified
//
#include <hip/hip_runtime.h>

#define N_PTS   16384
#define C_CH    24
#define N_GROUP 1024
#define GSIZE   32
#define B_SZ    16

// ---------------- FPS kernel: one 1024-thread block per batch ----------------
#define FPS_T 1024
#define PPT   (N_PTS / FPS_T)   // 16 points per thread, dists kept in registers

__global__ __launch_bounds__(FPS_T) void fps_kernel(const float* __restrict__ xyz,
                                                    float* __restrict__ center_out) {
    const int b = blockIdx.x;
    const int t = threadIdx.x;
    const float* xb = xyz + (size_t)b * N_PTS * C_CH;

    __shared__ float red_d[FPS_T];
    __shared__ int   red_i[FPS_T];
    __shared__ float cbuf[C_CH];

    float dists[PPT];
#pragma unroll
    for (int i = 0; i < PPT; ++i) dists[i] = __builtin_inff();

    int last = 0;
    for (int g = 0; g < N_GROUP; ++g) {
        // broadcast current center, also emit it (reference emits 'last' BEFORE update)
        if (t < C_CH) {
            float cv = xb[(size_t)last * C_CH + t];
            cbuf[t] = cv;
            center_out[((size_t)b * N_GROUP + g) * C_CH + t] = cv;
        }
        __syncthreads();

        float cc[C_CH];
#pragma unroll
        for (int c = 0; c < C_CH; ++c) cc[c] = cbuf[c];

        float best = -1.0f;
        int   bestn = 0;
#pragma unroll
        for (int i = 0; i < PPT; ++i) {
            const int n = t + i * FPS_T;
            const float* p = xb + (size_t)n * C_CH;
            float d = 0.0f;
#pragma unroll
            for (int c = 0; c < C_CH; ++c) { float df = p[c] - cc[c]; d += df * df; }
            float dv = fminf(dists[i], d);
            dists[i] = dv;
            if (dv > best) { best = dv; bestn = n; }   // ascending n => first-max tiebreak
        }
        red_d[t] = best;
        red_i[t] = bestn;
        __syncthreads();
        for (int s = FPS_T / 2; s > 0; s >>= 1) {
            if (t < s) {
                float d2v = red_d[t + s];
                int   i2  = red_i[t + s];
                if (d2v > red_d[t] || (d2v == red_d[t] && i2 < red_i[t])) {
                    red_d[t] = d2v; red_i[t] = i2;
                }
            }
            __syncthreads();
        }
        last = red_i[0];
        // next red_d write is separated from this read by the cbuf __syncthreads()
    }
}

// ------------- KNN(top-32 via f32 WMMA) + centered gather, fused -------------
// Workgroup = 256 threads (8 waves), owns a tile of 16 centers.
// Per iteration: async-stage 128 points (12KB) into LDS, then 8 waves each
// compute one 16x16 d2 tile with V_WMMA_F32_16X16X4_F32 (K=24 -> 6 WMMAs).

#define TILE_M  16
#define TILE_NW 16
#define NWAVES  8
#define TILE_N  (TILE_NW * NWAVES)   // 128
#define SLICES  16                   // threads maintaining top-k per center row

typedef float v2f __attribute__((ext_vector_type(2)));
typedef float v8f __attribute__((ext_vector_type(8)));

__global__ __launch_bounds__(256) void knn_group_kernel(const float* __restrict__ xyz,
                                                        const float* __restrict__ centers,
                                                        float* __restrict__ nbh_out) {
    const int b    = blockIdx.y;
    const int gt   = blockIdx.x;
    const int t    = threadIdx.x;
    const int lane = t & 31;
    const int wave = t >> 5;
    const int half = lane >> 4;    // 0: lanes 0-15, 1: lanes 16-31
    const int l16  = lane & 15;

    const float* xb = xyz + (size_t)b * N_PTS * C_CH;
    const float* cb = centers + ((size_t)b * N_GROUP + (size_t)gt * TILE_M) * C_CH;

    __shared__ __align__(16) float xtile[TILE_N][C_CH];   // async-staged xyz slab (12KB)
    __shared__ float cen[TILE_M][C_CH];
    __shared__ float cnorm[TILE_M];
    __shared__ float xnorm[TILE_N];
    __shared__ float d2buf[TILE_M][TILE_N];
    __shared__ float head_d[TILE_M][SLICES];
    __shared__ int   head_i[TILE_M][SLICES];
    __shared__ int   win_s[TILE_M];
    __shared__ int   topn[TILE_M][GSIZE];

    for (int e = t; e < TILE_M * C_CH; e += 256) cen[e / C_CH][e % C_CH] = cb[e];
    __syncthreads();
    if (t < TILE_M) {
        float s = 0.0f;
#pragma unroll
        for (int c = 0; c < C_CH; ++c) { float v = cen[t][c]; s += v * v; }
        cnorm[t] = s;
    }
    __syncthreads();

    // A fragments: 16x4 f32 per chunk; identical in every wave.
    v2f afrag[6];
#pragma unroll
    for (int j = 0; j < 6; ++j) {
        const int kk = 4 * j + 2 * half;
        afrag[j].x = cen[l16][kk];
        afrag[j].y = cen[l16][kk + 1];
    }

    // per-thread sorted top-32 over its column slice (registers, full unroll)
    const int row = t >> 4;
    const int sl  = t & 15;
    float kd[GSIZE];
    int   ki[GSIZE];
#pragma unroll
    for (int j = 0; j < GSIZE; ++j) { kd[j] = __builtin_inff(); ki[j] = 0; }

    const unsigned xtile_lds = (unsigned)(size_t)&xtile[0][0];  // LDS byte address
    const int w16l = wave * TILE_NW + l16;

    for (int nt = 0; nt < N_PTS; nt += TILE_N) {
        // --- async stage xyz[nt .. nt+128) into LDS: 768 x B128, 3 per thread ---
        {
            const unsigned long long gbase =
                (unsigned long long)(size_t)(xb + (size_t)nt * C_CH);
#pragma unroll
            for (int r = 0; r < 3; ++r) {
                const unsigned eo  = (unsigned)(t + r * 256) * 16u;  // byte offset
                const unsigned lza = xtile_lds + eo;
                asm volatile("global_load_async_to_lds_b128 %0, %1, %2 offset:0"
                             :: "v"(lza), "v"(eo), "s"(gbase)
                             : "memory");
            }
            asm volatile("s_wait_asynccnt 0x0" ::: "memory");
        }
        __syncthreads();   // xtile visible to all waves

        // point norms from the staged tile
        if (t < TILE_N) {
            float s = 0.0f;
#pragma unroll
            for (int c = 0; c < C_CH; ++c) { float v = xtile[t][c]; s += v * v; }
            xnorm[t] = s;
        }
        __syncthreads();

        // B fragments (4x16 f32 per chunk) from LDS, then the K=24 WMMA chain
        v8f acc = {};
#pragma unroll
        for (int j = 0; j < 6; ++j) {
            const int kk = 4 * j + 2 * half;
            v2f bfrag;
            bfrag.x = xtile[w16l][kk];
            bfrag.y = xtile[w16l][kk + 1];
            acc = __builtin_amdgcn_wmma_f32_16x16x4_f32(
                false, afrag[j], false, bfrag, (short)0, acc, false, false);
        }

        const float xn = xnorm[w16l];
#pragma unroll
        for (int r = 0; r < 8; ++r) {
            const int m = r + 8 * half;
            d2buf[m][w16l] = cnorm[m] + xn - 2.0f * acc[r];
        }
        __syncthreads();

        // insertion: thread (row, sl) consumes columns sl*8 .. sl*8+7
#pragma unroll
        for (int q = 0; q < 8; ++q) {
            const int   col = sl * 8 + q;
            const float d   = d2buf[row][col];
            const int   n   = nt + col;
            if (d < kd[GSIZE - 1]) {
                kd[GSIZE - 1] = d;
                ki[GSIZE - 1] = n;
#pragma unroll
                for (int j = GSIZE - 1; j > 0; --j) {
                    if (kd[j] < kd[j - 1]) {
                        float td = kd[j]; kd[j] = kd[j - 1]; kd[j - 1] = td;
                        int   ti = ki[j]; ki[j] = ki[j - 1]; ki[j - 1] = ti;
                    }
                }
            }
        }
        __syncthreads();   // also isolates xtile reuse next iteration
    }

    // 16-way streaming merge per row: 32 rounds, heads in LDS, lists shift in regs
    for (int out = 0; out < GSIZE; ++out) {
        head_d[row][sl] = kd[0];
        head_i[row][sl] = ki[0];
        __syncthreads();
        if (sl == 0) {
            float bd = head_d[row][0];
            int   bs = 0;
            for (int h = 1; h < SLICES; ++h) {
                float hd = head_d[row][h];
                if (hd < bd) { bd = hd; bs = h; }
            }
            topn[row][out] = head_i[row][bs];
            win_s[row]     = bs;
        }
        __syncthreads();
        if (sl == win_s[row]) {
#pragma unroll
            for (int j = 0; j < GSIZE - 1; ++j) { kd[j] = kd[j + 1]; ki[j] = ki[j + 1]; }
            kd[GSIZE - 1] = __builtin_inff();
        }
    }
    __syncthreads();

    // gather + center-subtract -> neighborhood [B,G,32,24]
    for (int e = t; e < TILE_M * GSIZE * C_CH; e += 256) {
        const int rr  = e / (GSIZE * C_CH);
        const int rem = e % (GSIZE * C_CH);
        const int m   = rem / C_CH;
        const int c   = rem % C_CH;
        const int n   = topn[rr][m];
        const float v = xb[(size_t)n * C_CH + c] - cen[rr][c];
        const size_t g = (size_t)gt * TILE_M + rr;
        nbh_out[(((size_t)b * N_GROUP + g) * GSIZE + m) * C_CH + c] = v;
    }
}

extern "C" void kernel_launch(void* const* d_in, const int* in_sizes, int n_in,
                              void* d_out, int out_size, void* d_ws, size_t ws_size,
                              hipStream_t stream) {
    (void)in_sizes; (void)n_in; (void)out_size; (void)d_ws; (void)ws_size;
    const float* xyz = (const float*)d_in[0];
    float* out = (float*)d_out;
    // d_out layout: neighborhood [16,1024,32,24] then center [16,1024,24]
    float* center_out = out + (size_t)B_SZ * N_GROUP * GSIZE * C_CH;

    fps_kernel<<<B_SZ, FPS_T, 0, stream>>>(xyz, center_out);

    dim3 grid2(N_GROUP / TILE_M, B_SZ);
    knn_group_kernel<<<grid2, 256, 0, stream>>>(xyz, center_out, out);
}